// UVRenderer_46256797778253
// MI455X (gfx1250) — compile-verified
//
#include <hip/hip_runtime.h>

// ---------------------------------------------------------------------------
// UV renderer (barycentric attribute interpolation) for MI455X / gfx1250.
//
// Reference collapses to: per pixel p with f = pix_to_face[p] >= 0,
//   out[p][d] = sum_k bary[p][k] * verts_attr[0][ face_tensor[f][k] ][d]
// broadcast identically over N=16 batches (all batches gather batch-0 attrs).
//
// Memory-bound: 201 MB output stores + ~16 MB streaming input @ 23.3 TB/s.
// Strategy:
//   * 82 KB vertex table staged into LDS via CDNA5 async global->LDS b128
//     copies (ASYNCcnt), persistent grid so staging amortizes.
//   * NT (non-temporal) loads/stores for the streaming pix/bary/output so the
//     192 MB L2 keeps the gather tables resident instead of thrashing on the
//     201 MB output stream.
//   * 4 pixels per thread -> everything moves as 16B-aligned b128.
// ---------------------------------------------------------------------------

typedef float v4f __attribute__((ext_vector_type(4)));
typedef int   v4i __attribute__((ext_vector_type(4)));
// GCC-style vector matching the async-builtin's parameter type:
typedef int   gv4i __attribute__((vector_size(16)));
typedef __attribute__((address_space(1))) gv4i* gv4i_global_ptr;
typedef __attribute__((address_space(3))) gv4i* gv4i_lds_ptr;

#define NUM_VERTS 6890
#define NUM_FACES 13776
#define NPIX      (1024 * 1024)
#define NBATCH    16
#define NGROUP    (NPIX / 4)          // 4 pixels per thread-iteration
#define HW3_V4    ((NPIX * 3) / 4)    // float4s per output batch
#define VERT_DW   (NUM_VERTS * 3)     // 20670 dwords in the vertex table
#define VERT_QW   ((VERT_DW + 3) / 4) // 5168 16B quads (8B overread stays in-bounds)

__global__ __launch_bounds__(256)
void uvrender_kernel(const float* __restrict__ verts,
                     const int*   __restrict__ faces,
                     const int*   __restrict__ pix,
                     const float* __restrict__ bary,
                     float*       __restrict__ out)
{
    __shared__ float s_verts[VERT_QW * 4];   // 82,688 bytes of LDS

    // ---- Stage verts_attr[0] into LDS with async global->LDS b128 copies ----
    for (int q = threadIdx.x; q < VERT_QW; q += 256) {
#if __has_builtin(__builtin_amdgcn_global_load_async_to_lds_b128)
        __builtin_amdgcn_global_load_async_to_lds_b128(
            (gv4i_global_ptr)(verts + 4 * (size_t)q),
            (gv4i_lds_ptr)(&s_verts[4 * q]),
            /*offset=*/0, /*cpol=*/0);
#else
        unsigned lofs = (unsigned)(unsigned long long)(&s_verts[4 * q]);
        const float* gsrc = verts + 4 * (size_t)q;
        asm volatile("global_load_async_to_lds_b128 %0, %1, off"
                     :: "v"(lofs), "v"(gsrc) : "memory");
#endif
    }
#if __has_builtin(__builtin_amdgcn_s_wait_asynccnt)
    __builtin_amdgcn_s_wait_asynccnt(0);
#else
    asm volatile("s_wait_asynccnt 0" ::: "memory");
#endif
    __syncthreads();

    // ---- Persistent grid-stride loop over groups of 4 pixels ----
    const int stride = gridDim.x * blockDim.x;
    for (int g = blockIdx.x * blockDim.x + threadIdx.x; g < NGROUP; g += stride) {
        // 4 face ids (read-once stream -> NT load, one b128)
        v4i p4 = __builtin_nontemporal_load((const v4i*)pix + g);
        // 12 barycentric floats for the 4 pixels (3 b128 NT loads)
        const v4f* bb = (const v4f*)bary + 3 * (size_t)g;
        v4f q0 = __builtin_nontemporal_load(bb + 0);
        v4f q1 = __builtin_nontemporal_load(bb + 1);
        v4f q2 = __builtin_nontemporal_load(bb + 2);

        float r[12];
#define INTERP(ri, fidx, b0, b1, b2)                                         \
        do {                                                                 \
            int f_ = (fidx);                                                 \
            if (f_ >= 0) {                                                   \
                const int* fp_ = faces + 3 * (size_t)f_;                     \
                int a_ = 3 * fp_[0];                                         \
                int b_ = 3 * fp_[1];                                         \
                int c_ = 3 * fp_[2];                                         \
                r[(ri)+0] = (b0)*s_verts[a_+0] + (b1)*s_verts[b_+0] + (b2)*s_verts[c_+0]; \
                r[(ri)+1] = (b0)*s_verts[a_+1] + (b1)*s_verts[b_+1] + (b2)*s_verts[c_+1]; \
                r[(ri)+2] = (b0)*s_verts[a_+2] + (b1)*s_verts[b_+2] + (b2)*s_verts[c_+2]; \
            } else {                                                         \
                r[(ri)+0] = 0.f; r[(ri)+1] = 0.f; r[(ri)+2] = 0.f;           \
            }                                                                \
        } while (0)

        INTERP(0, p4.x, q0.x, q0.y, q0.z);
        INTERP(3, p4.y, q0.w, q1.x, q1.y);
        INTERP(6, p4.z, q1.z, q1.w, q2.x);
        INTERP(9, p4.w, q2.y, q2.z, q2.w);
#undef INTERP

        v4f o0 = { r[0], r[1], r[2],  r[3]  };
        v4f o1 = { r[4], r[5], r[6],  r[7]  };
        v4f o2 = { r[8], r[9], r[10], r[11] };

        // Broadcast to all 16 batches; NT stores keep the 201 MB stream out
        // of L2 so the gather tables stay resident.
        v4f* ob = (v4f*)out + 3 * (size_t)g;
#pragma unroll
        for (int n = 0; n < NBATCH; ++n) {
            v4f* dst = ob + (size_t)n * HW3_V4;
            __builtin_nontemporal_store(o0, dst + 0);
            __builtin_nontemporal_store(o1, dst + 1);
            __builtin_nontemporal_store(o2, dst + 2);
        }
    }
}

extern "C" void kernel_launch(void* const* d_in, const int* in_sizes, int n_in,
                              void* d_out, int out_size, void* d_ws, size_t ws_size,
                              hipStream_t stream)
{
    (void)in_sizes; (void)n_in; (void)out_size; (void)d_ws; (void)ws_size;
    const float* verts = (const float*)d_in[0];   // (16, 6890, 3) f32
    const int*   faces = (const int*)  d_in[1];   // (13776, 3)   i32
    const int*   pix   = (const int*)  d_in[2];   // (1024, 1024) i32
    const float* bary  = (const float*)d_in[3];   // (1024,1024,3) f32
    float*       out   = (float*)d_out;           // (16,1024,1024,3) f32

    // 512 persistent blocks x 256 threads: 2 pixel-quads per thread; LDS
    // staging cost (512 x 82 KB = ~42 MB of L2-resident reads) amortized
    // against ~220 MB of streaming traffic.
    uvrender_kernel<<<dim3(512), dim3(256), 0, stream>>>(verts, faces, pix, bary, out);
}